// FrustumPointNetLoss_65472481460987
// MI455X (gfx1250) — compile-verified
//
#include <hip/hip_runtime.h>
#include <math.h>

#define NH 12
#define NS 6

#define NB1 2048   // blocks for mask-loss streaming reduction
#define T1  256
#define T2  256

typedef __attribute__((ext_vector_type(2))) float v2f;
typedef __attribute__((ext_vector_type(8))) float v8f;

__device__ __forceinline__ float huber_elem(float e, float delta) {
    float a = fabsf(e);
    float q = fminf(a, delta);
    return 0.5f * q * q + delta * (a - q);
}

// Exact f32 wave32 reduction using V_WMMA_F32_16X16X4_F32:
// A[16x4]: lane L (<16) supplies A[L][0..1] = {acc,0}; lane L+16 supplies A[L][2..3].
// B = all-ones (layout-irrelevant). D[m][n] = acc_m + acc_{m+16} for every n.
// Lane n<16 holds D[0..7][n]; lane n>=16 holds D[8..15][n-16]. Sum 8 regs, then
// add across lane^16 -> full 32-lane sum in every lane.
__device__ __forceinline__ float wave_reduce_sum(float acc) {
#if __has_builtin(__builtin_amdgcn_wmma_f32_16x16x4_f32)
    v2f a; a.x = acc;  a.y = 0.0f;
    v2f b; b.x = 1.0f; b.y = 1.0f;
    v8f c = {};
    c = __builtin_amdgcn_wmma_f32_16x16x4_f32(false, a, false, b, (short)0, c,
                                              false, false);
    float s = c[0] + c[1] + c[2] + c[3] + c[4] + c[5] + c[6] + c[7];
    s += __shfl_xor(s, 16, 32);
    return s;
#else
    for (int off = 16; off > 0; off >>= 1) acc += __shfl_xor(acc, off, 32);
    return acc;
#endif
}

// Stable NLL element for 2-way softmax: lse(s0,s1) - s[label]
__device__ __forceinline__ float nll2(float s0, float s1, int lbl) {
    float m = fmaxf(s0, s1);
    float d = fabsf(s0 - s1);
    float lse = m + log1pf(__expf(-d));
    float sel = lbl ? s1 : s0;
    return lse - sel;
}

// ---------------- Kernel 1: streaming mask-loss partial sums ----------------
__global__ void fpnl_mask_kernel(const float* __restrict__ logits,
                                 const int* __restrict__ labels,
                                 float* __restrict__ partials,
                                 long long n) {
    long long tid    = (long long)blockIdx.x * blockDim.x + threadIdx.x;
    long long stride = (long long)gridDim.x * blockDim.x;
    long long n4     = n >> 2;

    const float4* lg4 = (const float4*)logits;   // 2 elements per float4
    const int4*   lb4 = (const int4*)labels;     // 4 labels per int4

    float acc = 0.0f;
    for (long long i = tid; i < n4; i += stride) {
        __builtin_prefetch((const void*)(lg4 + 2 * (i + stride)), 0, 0);
        __builtin_prefetch((const void*)(lb4 + (i + stride)), 0, 0);
        float4 p0 = lg4[2 * i];
        float4 p1 = lg4[2 * i + 1];
        int4   lb = lb4[i];
        acc += nll2(p0.x, p0.y, lb.x);
        acc += nll2(p0.z, p0.w, lb.y);
        acc += nll2(p1.x, p1.y, lb.z);
        acc += nll2(p1.z, p1.w, lb.w);
    }
    for (long long e = (n4 << 2) + tid; e < n; e += stride) {
        acc += nll2(logits[2 * e], logits[2 * e + 1], labels[e]);
    }

    float wsum = wave_reduce_sum(acc);
    __shared__ float swave[T1 / 32];
    int lane = threadIdx.x & 31, wid = threadIdx.x >> 5;
    if (lane == 0) swave[wid] = wsum;
    __syncthreads();
    if (threadIdx.x == 0) {
        float s = 0.0f;
        for (int w = 0; w < T1 / 32; ++w) s += swave[w];
        partials[blockIdx.x] = s;
    }
}

// ---------------- Kernel 2: per-row losses (7 partial sums / block) ---------
__global__ void fpnl_row_kernel(const float* __restrict__ center,
                                const float* __restrict__ center_label,
                                const float* __restrict__ stage1_center,
                                const float* __restrict__ heading_scores,
                                const float* __restrict__ hrn,
                                const float* __restrict__ hres,
                                const int*   __restrict__ hcls_label,
                                const float* __restrict__ hres_label,
                                const float* __restrict__ size_scores,
                                const float* __restrict__ srn,
                                const float* __restrict__ sres,
                                const int*   __restrict__ scls_label,
                                const float* __restrict__ sres_label,
                                const float* __restrict__ anchors,
                                float* __restrict__ partials,
                                int bs) {
    const float PI = 3.14159265358979323846f;
    const float sxA[8] = {1.f, 1.f, -1.f, -1.f, 1.f, 1.f, -1.f, -1.f};
    const float syA[8] = {1.f, 1.f, 1.f, 1.f, -1.f, -1.f, -1.f, -1.f};
    const float szA[8] = {1.f, -1.f, -1.f, 1.f, 1.f, -1.f, -1.f, 1.f};

    int r = blockIdx.x * blockDim.x + threadIdx.x;
    float vals[7] = {0.f, 0.f, 0.f, 0.f, 0.f, 0.f, 0.f};

    if (r < bs) {
        float cx = center[3 * r + 0], cy = center[3 * r + 1], cz = center[3 * r + 2];
        float lx = center_label[3 * r + 0], ly = center_label[3 * r + 1], lz = center_label[3 * r + 2];
        float ax = stage1_center[3 * r + 0], ay = stage1_center[3 * r + 1], az = stage1_center[3 * r + 2];

        // center / stage1-center huber on L2 distance
        float dc = sqrtf((cx - lx) * (cx - lx) + (cy - ly) * (cy - ly) + (cz - lz) * (cz - lz));
        vals[0] = huber_elem(dc, 2.0f);
        float ds = sqrtf((cx - ax) * (cx - ax) + (cy - ay) * (cy - ay) + (cz - az) * (cz - az));
        vals[1] = huber_elem(ds, 1.0f);

        int hc = hcls_label[r];
        int sc = scls_label[r];

        // heading class NLL (12-way)
        {
            float m = -INFINITY;
            for (int h = 0; h < NH; ++h) m = fmaxf(m, heading_scores[r * NH + h]);
            float se = 0.f;
            for (int h = 0; h < NH; ++h) se += __expf(heading_scores[r * NH + h] - m);
            vals[2] = logf(se) + m - heading_scores[r * NH + hc];
        }
        // heading residual normalized huber
        {
            float target = hres_label[r] / (PI / (float)NH);
            vals[3] = huber_elem(hrn[r * NH + hc] - target, 1.0f);
        }
        // size class NLL (6-way)
        {
            float m = -INFINITY;
            for (int s = 0; s < NS; ++s) m = fmaxf(m, size_scores[r * NS + s]);
            float se = 0.f;
            for (int s = 0; s < NS; ++s) se += __expf(size_scores[r * NS + s] - m);
            vals[4] = logf(se) + m - size_scores[r * NS + sc];
        }
        // size residual normalized huber on L2 of 3-vector
        {
            float e2 = 0.f;
            for (int k = 0; k < 3; ++k) {
                float mean = anchors[sc * 3 + k];
                float labn = sres_label[3 * r + k] / mean;
                float pred = srn[(r * NS + sc) * 3 + k];
                float d = labn - pred;
                e2 += d * d;
            }
            vals[5] = huber_elem(sqrtf(e2), 1.0f);
        }
        // corners loss: selected (hc, sc) box vs gt and flipped gt
        {
            float bin = (float)hc * (2.0f * PI / (float)NH);
            float hp = hres[r * NH + hc] + bin;           // pred heading
            float hg = hres_label[r] + bin;               // gt heading
            float lp = anchors[sc * 3 + 0] + 2.f * sres[(r * NS + sc) * 3 + 0];
            float wp = anchors[sc * 3 + 1] + 2.f * sres[(r * NS + sc) * 3 + 1];
            float hhp = anchors[sc * 3 + 2] + 2.f * sres[(r * NS + sc) * 3 + 2];
            float lg = anchors[sc * 3 + 0] + sres_label[3 * r + 0];
            float wg = anchors[sc * 3 + 1] + sres_label[3 * r + 1];
            float hhg = anchors[sc * 3 + 2] + sres_label[3 * r + 2];
            float cp = __cosf(hp), sp = __sinf(hp);
            float cg = __cosf(hg), sg = __sinf(hg);
            float csum = 0.f;
            for (int j = 0; j < 8; ++j) {
                // pred corner (rotation about y)
                float xp = 0.5f * lp * sxA[j], yp = 0.5f * hhp * syA[j], zp = 0.5f * wp * szA[j];
                float px = cp * xp + sp * zp + cx;
                float py = yp + cy;
                float pz = -sp * xp + cp * zp + cz;
                // gt corner
                float xg = 0.5f * lg * sxA[j], yg = 0.5f * hhg * syA[j], zg = 0.5f * wg * szA[j];
                float gx = cg * xg + sg * zg + lx;
                float gy = yg + ly;
                float gz = -sg * xg + cg * zg + lz;
                // flipped gt (heading + pi -> cos,sin negate)
                float fx = -cg * xg - sg * zg + lx;
                float fz = sg * xg - cg * zg + lz;
                float d1 = sqrtf((px - gx) * (px - gx) + (py - gy) * (py - gy) + (pz - gz) * (pz - gz));
                float d2 = sqrtf((px - fx) * (px - fx) + (py - gy) * (py - gy) + (pz - fz) * (pz - fz));
                csum += huber_elem(fminf(d1, d2), 1.0f);
            }
            vals[6] = csum;
        }
    }

    __shared__ float swave[T2 / 32];
    int lane = threadIdx.x & 31, wid = threadIdx.x >> 5;
    for (int t = 0; t < 7; ++t) {
        float w = wave_reduce_sum(vals[t]);
        if (lane == 0) swave[wid] = w;
        __syncthreads();
        if (threadIdx.x == 0) {
            float s = 0.f;
            for (int k = 0; k < T2 / 32; ++k) s += swave[k];
            partials[blockIdx.x * 8 + t] = s;
        }
        __syncthreads();
    }
}

// ---------------- Kernel 3: deterministic finalize --------------------------
__global__ void fpnl_finalize_kernel(const float* __restrict__ mask_partials,
                                     const float* __restrict__ row_partials,
                                     float* __restrict__ out,
                                     long long n, int bs, int nb2) {
    __shared__ float sdata[256];
    float s = 0.f;
    for (int i = threadIdx.x; i < NB1; i += blockDim.x) s += mask_partials[i];
    sdata[threadIdx.x] = s;
    __syncthreads();
    for (int off = 128; off > 0; off >>= 1) {
        if (threadIdx.x < off) sdata[threadIdx.x] += sdata[threadIdx.x + off];
        __syncthreads();
    }
    if (threadIdx.x == 0) {
        float mask_loss = sdata[0] / (float)n;
        float sums[7] = {0.f, 0.f, 0.f, 0.f, 0.f, 0.f, 0.f};
        for (int b = 0; b < nb2; ++b)
            for (int t = 0; t < 7; ++t) sums[t] += row_partials[b * 8 + t];
        float inv = 1.0f / (float)bs;
        float center_loss  = sums[0] * inv;
        float stage1_loss  = sums[1] * inv;
        float hcls_loss    = sums[2] * inv;
        float hres_loss    = sums[3] * inv;
        float scls_loss    = sums[4] * inv;
        float sres_loss    = sums[5] * inv;
        float corners_loss = sums[6] / ((float)bs * 8.0f);
        float total = mask_loss + (center_loss + hcls_loss + scls_loss +
                                   hres_loss * 20.f + sres_loss * 20.f +
                                   stage1_loss + 10.f * corners_loss);
        out[0] = total;
        out[1] = mask_loss;
        out[2] = center_loss;
        out[3] = hcls_loss;
        out[4] = scls_loss;
        out[5] = 20.f * hres_loss;
        out[6] = 20.f * sres_loss;
        out[7] = stage1_loss;
        out[8] = 10.f * corners_loss;
    }
}

extern "C" void kernel_launch(void* const* d_in, const int* in_sizes, int n_in,
                              void* d_out, int out_size, void* d_ws, size_t ws_size,
                              hipStream_t stream) {
    const float* logits        = (const float*)d_in[0];
    const int*   mask_label    = (const int*)d_in[1];
    const float* center        = (const float*)d_in[2];
    const float* center_label  = (const float*)d_in[3];
    const float* stage1_center = (const float*)d_in[4];
    const float* heading_scores= (const float*)d_in[5];
    const float* hrn           = (const float*)d_in[6];
    const float* hres          = (const float*)d_in[7];
    const int*   hcl           = (const int*)d_in[8];
    const float* hrl           = (const float*)d_in[9];
    const float* size_scores   = (const float*)d_in[10];
    const float* srn           = (const float*)d_in[11];
    const float* sres          = (const float*)d_in[12];
    const int*   scl           = (const int*)d_in[13];
    const float* srl           = (const float*)d_in[14];
    const float* anchors       = (const float*)d_in[15];

    long long n  = (long long)in_sizes[1]; // BS*NP elements for mask loss
    int       bs = in_sizes[2] / 3;        // BS

    float* ws            = (float*)d_ws;
    float* mask_partials = ws;             // NB1 floats
    float* row_partials  = ws + NB1;       // nb2*8 floats
    int nb2 = (bs + T2 - 1) / T2;

    fpnl_mask_kernel<<<NB1, T1, 0, stream>>>(logits, mask_label, mask_partials, n);
    fpnl_row_kernel<<<nb2, T2, 0, stream>>>(center, center_label, stage1_center,
                                            heading_scores, hrn, hres, hcl, hrl,
                                            size_scores, srn, sres, scl, srl,
                                            anchors, row_partials, bs);
    fpnl_finalize_kernel<<<1, 256, 0, stream>>>(mask_partials, row_partials,
                                                (float*)d_out, n, bs, nb2);
}